// FactorizedEmbedding_47837345743217
// MI455X (gfx1250) — compile-verified
//
#include <hip/hip_runtime.h>

typedef __attribute__((ext_vector_type(2))) float v2f;
typedef __attribute__((ext_vector_type(4))) float v4f;
typedef __attribute__((ext_vector_type(8))) float v8f;

// Fused gather-GEMM:  out[t, :] = embed_vocab[x[t], :] @ embed_hidden
//   A  : gathered rows [n_tokens, E]   (E = 128)
//   W  : embed_hidden  [E, H]          (H = 1024, row-major, L2-resident 512KB)
//   OUT: [n_tokens, H] fp32 (streamed once, nontemporal)
//
// Block = 256 threads = 8 wave32, covering a 64-token strip (4 row-tiles).
// A-strip staged in LDS (stride 132 -> conflict-free ds_load_b64 fragments).
// Each wave walks 16-wide column tiles; per K-step it loads ONE B fragment
// and feeds FOUR v_wmma_f32_16x16x4_f32 (4 row-tiles), quartering W traffic.

#define WAVES       8
#define ROW_TILES   4
#define STRIP       (ROW_TILES * 16)   // 64 tokens per block
#define MAX_E       128
#define LDS_STRIDE  (MAX_E + 4)        // 132: +4 floats -> 4-bank shift per row

template <int E_, int H_>
__global__ __launch_bounds__(256) void
factorized_embed_wmma(const int* __restrict__ x,
                      const float* __restrict__ Ev,   // [V, E]
                      const float* __restrict__ Eh,   // [E, H]
                      float* __restrict__ out,        // [n_tokens, H]
                      int Er, int Hr, int n_tokens)
{
    const int E = E_ ? E_ : Er;        // compile-time for the hot instantiation
    const int H = H_ ? H_ : Hr;

    __shared__ float ldsA[STRIP * LDS_STRIDE];        // 64 x 132 fp32 = 33 KB

    const int tid     = threadIdx.x;
    const int lane    = tid & 31;
    const int wave    = tid >> 5;
    const int rowBase = blockIdx.x * STRIP;

    // ---- cooperative gather of 64 embedding rows into LDS ----------------
    const int chunksPerRow = E >> 3;                  // 8 floats per chunk
    for (int i = tid; i < STRIP * chunksPerRow; i += 256) {
        const int r = i / chunksPerRow;
        const int c = (i - r * chunksPerRow) << 3;
        int row = rowBase + r;
        if (row >= n_tokens) row = n_tokens - 1;      // clamp padding rows
        const int tok = x[row];
        const float* src = Ev + (size_t)tok * (size_t)E + c;
        *(v4f*)(&ldsA[r * LDS_STRIDE + c])     = *(const v4f*)(src);
        *(v4f*)(&ldsA[r * LDS_STRIDE + c + 4]) = *(const v4f*)(src + 4);
    }
    __syncthreads();

    // ---- WMMA fragment lane mapping (ISA 7.12.2, wave32) -----------------
    const int laneN  = lane & 15;      // A: row M,  B/C/D: col N
    const int laneHi = lane >> 4;      // 0: low K/M half, 1: high half

    const int numColTiles = H >> 4;
    const int kSteps      = E >> 2;

    for (int ct = blockIdx.y * WAVES + wave; ct < numColTiles;
         ct += gridDim.y * WAVES) {
        const int colBase = ct << 4;

        v8f acc[ROW_TILES] = {};

        // B column pointer: W[laneHi*2][colBase + laneN]; advances 4 rows/step
        const float* bptr = Eh + (size_t)(laneHi << 1) * (size_t)H
                               + (size_t)(colBase + laneN);
        // A fragment pointer: row laneN, col laneHi*2; advances 4 cols/step
        const float* aptr = &ldsA[laneN * LDS_STRIDE + (laneHi << 1)];

        #pragma unroll 8
        for (int kk = 0; kk < kSteps; ++kk) {
            v2f b;
            b.x = bptr[0];
            b.y = bptr[(size_t)H];

            #pragma unroll
            for (int rt = 0; rt < ROW_TILES; ++rt) {
                v2f a = *(const v2f*)(aptr + rt * 16 * LDS_STRIDE);
                acc[rt] = __builtin_amdgcn_wmma_f32_16x16x4_f32(
                              /*neg_a=*/false, a,
                              /*neg_b=*/false, b,
                              /*c_mod=*/(short)0, acc[rt],
                              /*reuse_a=*/false, /*reuse_b=*/false);
            }
            bptr += (size_t)4 * (size_t)H;
            aptr += 4;
        }

        // ---- store: VGPR r of tile rt -> row (rt*16 + r + 8*laneHi) ------
        #pragma unroll
        for (int rt = 0; rt < ROW_TILES; ++rt) {
            #pragma unroll
            for (int r = 0; r < 8; ++r) {
                const int row = rowBase + rt * 16 + r + (laneHi << 3);
                if (row < n_tokens) {
                    __builtin_nontemporal_store(
                        acc[rt][r],
                        &out[(size_t)row * (size_t)H + colBase + laneN]);
                }
            }
        }
    }
}

extern "C" void kernel_launch(void* const* d_in, const int* in_sizes, int n_in,
                              void* d_out, int out_size, void* d_ws, size_t ws_size,
                              hipStream_t stream)
{
    const int*   x   = (const int*)  d_in[0];   // [B*S] token ids
    const float* Ev  = (const float*)d_in[1];   // [V, E]
    const float* Eh  = (const float*)d_in[2];   // [E, H]
    float*       out = (float*)d_out;           // [B*S, H]

    const int n_tokens = in_sizes[0];           // 16384
    const int H        = out_size / n_tokens;   // 1024
    const int E        = in_sizes[2] / H;       // 128

    dim3 grid((n_tokens + STRIP - 1) / STRIP,   // 256 row strips
              (H >= 512) ? 2 : 1);              // x2 column split -> 4096 waves

    if (E == 128 && H == 1024) {
        factorized_embed_wmma<128, 1024>
            <<<grid, 256, 0, stream>>>(x, Ev, Eh, out, E, H, n_tokens);
    } else {
        factorized_embed_wmma<0, 0>
            <<<grid, 256, 0, stream>>>(x, Ev, Eh, out, E, H, n_tokens);
    }
}